// Model_86586540687792
// MI455X (gfx1250) — compile-verified
//
#include <hip/hip_runtime.h>
#include <hip/hip_bf16.h>

typedef __attribute__((ext_vector_type(2))) float v2f;
typedef __attribute__((ext_vector_type(8))) float v8f;

#define B_      2
#define C_      4
#define N_      200000
#define M_      25000                      // mask words per (b,c): 8 bits each
#define TILE    256
#define NTILES  ((N_ + TILE - 1) / TILE)   // 782
#define ROWS_PAD ((NTILES + 3) & ~3)       // 784 (multiple of 4 for K=4 WMMA)
#define NB1     ((M_ + 255) / 256)         // 98 scan blocks per (b,c)

// ---------------------------------------------------------------------------
// Scan kernel 1: per-word popcounts, block-level exclusive scan.
// grid = (NB1, B*C), block = 256
// ---------------------------------------------------------------------------
__global__ void scan_words(const int* __restrict__ mask,
                           int* __restrict__ wordExcl,
                           int* __restrict__ blockTot) {
    __shared__ int s[256];
    const int bc = blockIdx.y;
    const int t  = threadIdx.x;
    const int w  = blockIdx.x * 256 + t;
    int pc = 0;
    if (w < M_) pc = __popc(mask[bc * M_ + w] & 0xFF);
    s[t] = pc;
    __syncthreads();
#pragma unroll
    for (int off = 1; off < 256; off <<= 1) {
        int u = (t >= off) ? s[t - off] : 0;
        __syncthreads();
        s[t] += u;
        __syncthreads();
    }
    if (w < M_) wordExcl[bc * M_ + w] = s[t] - pc;
    if (t == 255) blockTot[bc * NB1 + blockIdx.x] = s[255];
}

// ---------------------------------------------------------------------------
// Scan kernel 2: exclusive scan of the NB1 block totals per (b,c).
// Also zeroes the WMMA pad rows (NTILES..ROWS_PAD-1) of the partials buffer,
// so the WMMA reduction kernel can load unconditionally (no EXEC divergence).
// grid = (1, B*C), block = 128  (NB1 = 98 <= 128)
// ---------------------------------------------------------------------------
__global__ void scan_blocks(const int* __restrict__ blockTot,
                            int* __restrict__ blockExcl,
                            float* __restrict__ partials) {
    __shared__ int s[128];
    const int bc = blockIdx.y;
    const int t  = threadIdx.x;
    int v = (t < NB1) ? blockTot[bc * NB1 + t] : 0;
    s[t] = v;
    __syncthreads();
#pragma unroll
    for (int off = 1; off < 128; off <<= 1) {
        int u = (t >= off) ? s[t - off] : 0;
        __syncthreads();
        s[t] += u;
        __syncthreads();
    }
    if (t < NB1) blockExcl[bc * NB1 + t] = s[t] - v;
    // zero pad rows: (ROWS_PAD - NTILES) rows * 16 cols = 32 floats per (b,c)
    const int npad = (ROWS_PAD - NTILES) * 16;
    if (t < npad)
        partials[((size_t)bc * ROWS_PAD + NTILES) * 16 + t] = 0.0f;
}

// ---------------------------------------------------------------------------
// Main backward kernel: one thread per gaussian, loops over C cameras.
// grid = (NTILES, B), block = 256 (8 wave32)
// Emits per-block partial v_Rview rows (9 values padded to 16 cols) to ws.
// ---------------------------------------------------------------------------
__global__ void __launch_bounds__(256)
gsplat_bwd(const float* __restrict__ means,     // (B,3,N)
           const float* __restrict__ quats,     // (B,4,N)
           const float* __restrict__ scales,    // (B,3,N)
           const float* __restrict__ conics,    // (B,C,3,N)
           const float* __restrict__ viewmats,  // (B,C,4,4)
           const float* __restrict__ Ks,        // (B,C,3,3)
           const float* __restrict__ v_m2d,     // (B,C,2,N)
           const float* __restrict__ v_dep,     // (B,C,N)
           const float* __restrict__ v_con,     // (B,C,3,N)
           const float* __restrict__ v_col,     // (B,C,3,N)
           const float* __restrict__ v_op,      // (B,C,N)
           const int*   __restrict__ mask,      // (B,C,M)
           const int*   __restrict__ widthp,
           const int*   __restrict__ heightp,
           const int*   __restrict__ wordExcl,  // (B*C, M)
           const int*   __restrict__ blockExcl, // (B*C, NB1)
           float* __restrict__ partials,        // (B*C, ROWS_PAD, 16)
           float* __restrict__ out) {
    __shared__ float s_rv[C_ * 9];

    const int tid  = threadIdx.x;
    const int lane = tid & 31;
    const int b    = blockIdx.y;
    const int nRaw = blockIdx.x * TILE + tid;
    const bool valid = (nRaw < N_);
    const int n = valid ? nRaw : (N_ - 1);

    if (tid < C_ * 9) s_rv[tid] = 0.0f;
    __syncthreads();

    const float W = (float)widthp[0];
    const float H = (float)heightp[0];

    // ---- per-gaussian geometry -------------------------------------------
    float mean[3], q[4], sc[3];
#pragma unroll
    for (int i = 0; i < 3; ++i) mean[i] = means[((size_t)b * 3 + i) * N_ + n];
#pragma unroll
    for (int i = 0; i < 4; ++i) q[i] = quats[((size_t)b * 4 + i) * N_ + n];
#pragma unroll
    for (int i = 0; i < 3; ++i) sc[i] = scales[((size_t)b * 3 + i) * N_ + n];

    const float inv_norm = rsqrtf(q[0]*q[0] + q[1]*q[1] + q[2]*q[2] + q[3]*q[3]);
    const float qw = q[0]*inv_norm, qx = q[1]*inv_norm, qy = q[2]*inv_norm, qz = q[3]*inv_norm;

    float R[3][3];
    R[0][0] = 1.f - 2.f*(qy*qy + qz*qz); R[0][1] = 2.f*(qx*qy - qw*qz); R[0][2] = 2.f*(qx*qz + qw*qy);
    R[1][0] = 2.f*(qx*qy + qw*qz); R[1][1] = 1.f - 2.f*(qx*qx + qz*qz); R[1][2] = 2.f*(qy*qz - qw*qx);
    R[2][0] = 2.f*(qx*qz - qw*qy); R[2][1] = 2.f*(qy*qz + qw*qx); R[2][2] = 1.f - 2.f*(qx*qx + qy*qy);

    float Mm[3][3], cov[3][3];
#pragma unroll
    for (int i = 0; i < 3; ++i)
#pragma unroll
        for (int j = 0; j < 3; ++j) Mm[i][j] = R[i][j] * sc[j];
#pragma unroll
    for (int i = 0; i < 3; ++i)
#pragma unroll
        for (int k = 0; k < 3; ++k)
            cov[i][k] = Mm[i][0]*Mm[k][0] + Mm[i][1]*Mm[k][1] + Mm[i][2]*Mm[k][2];

    float vpW[3]  = {0.f, 0.f, 0.f};
    float covW[3][3] = {{0}};
    float vcol[3] = {0.f, 0.f, 0.f};
    float vop     = 0.f;

    const int wi  = n >> 3;
    const int bit = n & 7;

    // ---- camera loop ------------------------------------------------------
    for (int c = 0; c < C_; ++c) {
        const int bc = b * C_ + c;
        const float* Vm = viewmats + (size_t)bc * 16;
        const float* Km = Ks + (size_t)bc * 9;
        float Rv[3][3], tv[3];
#pragma unroll
        for (int i = 0; i < 3; ++i) {
#pragma unroll
            for (int j = 0; j < 3; ++j) Rv[i][j] = Vm[i*4 + j];
            tv[i] = Vm[i*4 + 3];
        }
        const float fx = Km[0], fy = Km[4], cx = Km[2], cy = Km[5];

        // compaction rank
        const int word = mask[(size_t)bc * M_ + wi];
        const bool act = ((word >> bit) & 1) != 0;
        const int rank = blockExcl[bc * NB1 + (wi >> 8)] + wordExcl[(size_t)bc * M_ + wi]
                       + __popc(word & ((1 << bit) - 1));

        float vc0 = 0.f, vc1 = 0.f, vc2 = 0.f, vm0 = 0.f, vm1 = 0.f, vdp = 0.f;
        float vl0 = 0.f, vl1 = 0.f, vl2 = 0.f, vo = 0.f;
        if (act) {
            const size_t b3 = (size_t)bc * 3 * N_;
            vc0 = v_con[b3 + 0*N_ + rank];
            vc1 = v_con[b3 + 1*N_ + rank];
            vc2 = v_con[b3 + 2*N_ + rank];
            vm0 = v_m2d[((size_t)bc * 2 + 0) * N_ + rank];
            vm1 = v_m2d[((size_t)bc * 2 + 1) * N_ + rank];
            vdp = v_dep[(size_t)bc * N_ + rank];
            vl0 = v_col[b3 + 0*N_ + rank];
            vl1 = v_col[b3 + 1*N_ + rank];
            vl2 = v_col[b3 + 2*N_ + rank];
            vo  = v_op[(size_t)bc * N_ + rank];
        }
        vcol[0] += vl0; vcol[1] += vl1; vcol[2] += vl2; vop += vo;

        // v_covar2d = -Ainv @ v_ci @ Ainv   (Ainv = conic 2x2, v_ci symmetric)
        const size_t cb = (size_t)bc * 3 * N_;
        const float a11 = conics[cb + 0*N_ + n];
        const float a12 = conics[cb + 1*N_ + n];
        const float a22 = conics[cb + 2*N_ + n];
        const float h = 0.5f * vc1;
        const float T00 = a11*vc0 + a12*h,  T01 = a11*h + a12*vc2;
        const float T10 = a12*vc0 + a22*h,  T11 = a12*h + a22*vc2;
        const float vC00 = -(T00*a11 + T01*a12);
        const float vC01 = -(T00*a12 + T01*a22);
        const float vC10 = -(T10*a11 + T11*a12);
        const float vC11 = -(T10*a12 + T11*a22);

        // camera-space mean + covariance
        const float mx = Rv[0][0]*mean[0] + Rv[0][1]*mean[1] + Rv[0][2]*mean[2] + tv[0];
        const float my = Rv[1][0]*mean[0] + Rv[1][1]*mean[1] + Rv[1][2]*mean[2] + tv[1];
        const float tz = Rv[2][0]*mean[0] + Rv[2][1]*mean[1] + Rv[2][2]*mean[2] + tv[2];
        const float tz2 = tz*tz, tz3 = tz2*tz;

        float tmp[3][3], cc[3][3];
#pragma unroll
        for (int i = 0; i < 3; ++i)
#pragma unroll
            for (int k = 0; k < 3; ++k)
                tmp[i][k] = Rv[i][0]*cov[0][k] + Rv[i][1]*cov[1][k] + Rv[i][2]*cov[2][k];
#pragma unroll
        for (int i = 0; i < 3; ++i)
#pragma unroll
            for (int l = 0; l < 3; ++l)
                cc[i][l] = tmp[i][0]*Rv[l][0] + tmp[i][1]*Rv[l][1] + tmp[i][2]*Rv[l][2];

        const float tanx = 0.5f * W / fx, tany = 0.5f * H / fy;
        const float lxp = (W - cx)/fx + 0.3f*tanx, lxn = cx/fx + 0.3f*tanx;
        const float lyp = (H - cy)/fy + 0.3f*tany, lyn = cy/fy + 0.3f*tany;
        const float rx = mx / tz, ry = my / tz;
        const float xf = (rx <= lxp && rx >= -lxn) ? 1.f : 0.f;
        const float yf = (ry <= lyp && ry >= -lyn) ? 1.f : 0.f;
        const float tx = tz * fminf(fmaxf(rx, -lxn), lxp);
        const float ty = tz * fminf(fmaxf(ry, -lyn), lyp);

        const float J00 = fx / tz, J02 = -fx * tx / tz2;
        const float J11 = fy / tz, J12 = -fy * ty / tz2;

        // G = J^T vC J  (3x3 symmetric)
        const float U00 = vC00*J00, U01 = vC01*J11, U02 = vC00*J02 + vC01*J12;
        const float U10 = vC10*J00, U11 = vC11*J11, U12 = vC10*J02 + vC11*J12;
        float G[3][3];
        G[0][0] = J00*U00; G[0][1] = J00*U01; G[0][2] = J00*U02;
        G[1][0] = J11*U10; G[1][1] = J11*U11; G[1][2] = J11*U12;
        G[2][0] = J02*U00 + J12*U10; G[2][1] = J02*U01 + J12*U11; G[2][2] = J02*U02 + J12*U12;

        // v_mean_c
        float m0 = J00 * vm0;
        float m1 = J11 * vm1;
        float m2 = -(fx*mx*vm0 + fy*my*vm1) / tz2;

        // v_J = vC@J@cc + vC^T@J@cc  (cc symmetric)
        const float S00 = U00 + vC00*J00, S01 = U01 + vC10*J11, S02 = U02 + (vC00*J02 + vC10*J12);
        const float S10 = U10 + vC01*J00, S11 = U11 + vC11*J11, S12 = U12 + (vC01*J02 + vC11*J12);
        const float vJ00 = S00*cc[0][0] + S01*cc[0][1] + S02*cc[0][2];
        const float vJ02 = S00*cc[2][0] + S01*cc[2][1] + S02*cc[2][2];
        const float vJ11 = S10*cc[1][0] + S11*cc[1][1] + S12*cc[1][2];
        const float vJ12 = S10*cc[2][0] + S11*cc[2][1] + S12*cc[2][2];

        m0 -= fx / tz2 * vJ02 * xf;
        m2 -= fx / tz3 * vJ02 * tx * (1.f - xf);
        m1 -= fy / tz2 * vJ12 * yf;
        m2 -= fy / tz3 * vJ12 * ty * (1.f - yf);
        m2 += -fx/tz2*vJ00 - fy/tz2*vJ11 + 2.f*fx*tx/tz3*vJ02 + 2.f*fy*ty/tz3*vJ12;
        m2 += vdp;

        // v_pW += Rv^T @ v_mean_c
        vpW[0] += Rv[0][0]*m0 + Rv[1][0]*m1 + Rv[2][0]*m2;
        vpW[1] += Rv[0][1]*m0 + Rv[1][1]*m1 + Rv[2][1]*m2;
        vpW[2] += Rv[0][2]*m0 + Rv[1][2]*m1 + Rv[2][2]*m2;

        // Z = G @ Rv
        float Z[3][3];
#pragma unroll
        for (int i = 0; i < 3; ++i)
#pragma unroll
            for (int k = 0; k < 3; ++k)
                Z[i][k] = G[i][0]*Rv[0][k] + G[i][1]*Rv[1][k] + G[i][2]*Rv[2][k];
        // covW += Rv^T @ Z
#pragma unroll
        for (int j = 0; j < 3; ++j)
#pragma unroll
            for (int l = 0; l < 3; ++l)
                covW[j][l] += Rv[0][j]*Z[0][l] + Rv[1][j]*Z[1][l] + Rv[2][j]*Z[2][l];

        // rv = v_mean_c (x) mean + 2 * Z @ cov  -> reduce to per-block partial
        const float mc[3] = {m0, m1, m2};
        const float fval = valid ? 1.f : 0.f;
#pragma unroll
        for (int i = 0; i < 3; ++i) {
#pragma unroll
            for (int l = 0; l < 3; ++l) {
                float v = mc[i]*mean[l]
                        + 2.f*(Z[i][0]*cov[0][l] + Z[i][1]*cov[1][l] + Z[i][2]*cov[2][l]);
                v *= fval;
                v += __shfl_down(v, 16, 32);
                v += __shfl_down(v,  8, 32);
                v += __shfl_down(v,  4, 32);
                v += __shfl_down(v,  2, 32);
                v += __shfl_down(v,  1, 32);
                if (lane == 0) atomicAdd(&s_rv[c*9 + i*3 + l], v);
            }
        }
    }

    // ---- post-camera per-gaussian gradients ------------------------------
    float vM[3][3];
#pragma unroll
    for (int i = 0; i < 3; ++i)
#pragma unroll
        for (int j = 0; j < 3; ++j)
            vM[i][j] = (covW[i][0] + covW[0][i]) * Mm[0][j]
                     + (covW[i][1] + covW[1][i]) * Mm[1][j]
                     + (covW[i][2] + covW[2][i]) * Mm[2][j];

    float vRr[3][3];
#pragma unroll
    for (int i = 0; i < 3; ++i)
#pragma unroll
        for (int j = 0; j < 3; ++j) vRr[i][j] = vM[i][j] * sc[j];

    const float a_ = vRr[2][1] - vRr[1][2];
    const float b_ = vRr[0][2] - vRr[2][0];
    const float c_ = vRr[1][0] - vRr[0][1];
    const float s1122 = vRr[1][1] + vRr[2][2];
    const float s0022 = vRr[0][0] + vRr[2][2];
    const float s0011 = vRr[0][0] + vRr[1][1];
    const float p01 = vRr[1][0] + vRr[0][1];
    const float p02 = vRr[2][0] + vRr[0][2];
    const float p12 = vRr[2][1] + vRr[1][2];
    float vq[4];
    vq[0] = 2.f*(qx*a_ + qy*b_ + qz*c_);
    vq[1] = 2.f*(-2.f*qx*s1122 + qy*p01 + qz*p02 + qw*a_);
    vq[2] = 2.f*(qx*p01 - 2.f*qy*s0022 + qz*p12 + qw*b_);
    vq[3] = 2.f*(qx*p02 + qy*p12 - 2.f*qz*s0011 + qw*c_);
    const float dotq = vq[0]*qw + vq[1]*qx + vq[2]*qy + vq[3]*qz;
    const float qn4[4] = {qw, qx, qy, qz};

    // ---- stores -----------------------------------------------------------
    if (valid) {
        const size_t offQ = (size_t)B_*N_*3;
        const size_t offS = (size_t)B_*N_*7;
        const size_t offC = (size_t)B_*N_*10 + B_*C_*9;
        const size_t offO = offC + (size_t)B_*3*N_;
#pragma unroll
        for (int j = 0; j < 3; ++j)
            out[(size_t)b*N_*3 + (size_t)n*3 + j] = vpW[j];
#pragma unroll
        for (int j = 0; j < 4; ++j)
            out[offQ + (size_t)b*N_*4 + (size_t)n*4 + j] = (vq[j] - dotq*qn4[j]) * inv_norm;
#pragma unroll
        for (int j = 0; j < 3; ++j)
            out[offS + (size_t)b*N_*3 + (size_t)n*3 + j] =
                R[0][j]*vM[0][j] + R[1][j]*vM[1][j] + R[2][j]*vM[2][j];
#pragma unroll
        for (int j = 0; j < 3; ++j)
            out[offC + ((size_t)b*3 + j)*N_ + n] = vcol[j];
        out[offO + (size_t)b*N_ + n] = vop;
    }

    __syncthreads();
    if (tid < 64) {
        const int c = tid >> 4, col = tid & 15;
        const float v = (col < 9) ? s_rv[c*9 + col] : 0.f;
        partials[(((size_t)(b*C_ + c))*ROWS_PAD + blockIdx.x)*16 + col] = v;
    }
}

// ---------------------------------------------------------------------------
// v_Rview final reduction via V_WMMA_F32_16X16X4_F32.
// A = ones(16x4)  ->  D[m][n] = sum_k B[k][n]  (column sums of 4 partial rows)
// Pad rows are pre-zeroed, so loads are unconditional (EXEC stays all-ones,
// no divergence around the WMMA chain). One wave per (b,c).
// grid = (B*C), block = 32
// ---------------------------------------------------------------------------
__global__ void rview_wmma_reduce(const float* __restrict__ partials,
                                  float* __restrict__ out_rv) {
    const int bc   = blockIdx.x;
    const int lane = threadIdx.x;
    const int col  = lane & 15;
    const int half = lane >> 4;
    const float* base = partials + (size_t)bc * ROWS_PAD * 16 + col;

    v8f acc = {0.f, 0.f, 0.f, 0.f, 0.f, 0.f, 0.f, 0.f};
    v2f a = {1.0f, 1.0f};   // all-ones A: layout-independent column reduction

    for (int chunk = 0; chunk < ROWS_PAD / 4; ++chunk) {
        const int r0 = chunk*4 + half;       // rows {half, 2+half} of this chunk
        v2f bv;
        bv[0] = base[r0*16];
        bv[1] = base[(r0 + 2)*16];
        acc = __builtin_amdgcn_wmma_f32_16x16x4_f32(
            false, a, false, bv, (short)0, acc, false, false);
    }
    // D row 0: VGPR0 of lanes 0..15 holds [M=0][N=lane]; cols 0..8 are v_Rview.
    if (lane < 9) out_rv[bc*9 + lane] = acc[0];
}

// ---------------------------------------------------------------------------
extern "C" void kernel_launch(void* const* d_in, const int* in_sizes, int n_in,
                              void* d_out, int out_size, void* d_ws, size_t ws_size,
                              hipStream_t stream) {
    const float* means    = (const float*)d_in[0];
    const float* quats    = (const float*)d_in[1];
    const float* scales   = (const float*)d_in[2];
    const float* conics   = (const float*)d_in[3];
    const float* viewmats = (const float*)d_in[4];
    const float* Ks       = (const float*)d_in[5];
    const float* v_m2d    = (const float*)d_in[6];
    const float* v_dep    = (const float*)d_in[7];
    const float* v_con    = (const float*)d_in[8];
    const float* v_col    = (const float*)d_in[9];
    const float* v_op     = (const float*)d_in[10];
    const int*   mask     = (const int*)d_in[11];
    const int*   widthp   = (const int*)d_in[12];
    const int*   heightp  = (const int*)d_in[13];

    char* ws = (char*)d_ws;
    const size_t SZ_WORD = (size_t)B_ * C_ * M_ * sizeof(int);    // 800000
    const size_t SZ_BLK  = (size_t)B_ * C_ * NB1 * sizeof(int);   // 3136
    int*   wordExcl  = (int*)ws;
    int*   blockTot  = (int*)(ws + SZ_WORD);
    int*   blockExcl = (int*)(ws + SZ_WORD + SZ_BLK);
    float* partials  = (float*)(ws + SZ_WORD + 2*SZ_BLK);

    float* out = (float*)d_out;
    float* out_rv = out + (size_t)B_ * N_ * 10;   // v_Rview offset in flat tuple

    scan_words<<<dim3(NB1, B_*C_), 256, 0, stream>>>(mask, wordExcl, blockTot);
    scan_blocks<<<dim3(1, B_*C_), 128, 0, stream>>>(blockTot, blockExcl, partials);
    gsplat_bwd<<<dim3(NTILES, B_), 256, 0, stream>>>(
        means, quats, scales, conics, viewmats, Ks,
        v_m2d, v_dep, v_con, v_col, v_op, mask, widthp, heightp,
        wordExcl, blockExcl, partials, out);
    rview_wmma_reduce<<<B_*C_, 32, 0, stream>>>(partials, out_rv);
}